// DeepSAT_27144193311172
// MI455X (gfx1250) — compile-verified
//
#include <hip/hip_runtime.h>
#include <hip/hip_bf16.h>

typedef __attribute__((ext_vector_type(2))) float v2f;
typedef __attribute__((ext_vector_type(8))) float v8f;

constexpr int HD    = 128;   // hidden dim
constexpr int DEPTH = 8;
constexpr int AST   = 132;   // padded LDS row stride (conflict-free column reads)
constexpr int MB    = 32;    // nodes (M rows) per block: 2 row-tiles per B fragment

// ---------------------------------------------------------------------------
// order-preserving float atomic max (values init to -1e9)
__device__ __forceinline__ void atomicMaxFloat(float* addr, float val) {
    if (val >= 0.0f) atomicMax((int*)addr, __float_as_int(val));
    else             atomicMin((unsigned int*)addr, __float_as_uint(val));
}

// ---------------------------------------------------------------------------
__global__ void reset_kernel(float* __restrict__ agg, float* __restrict__ m,
                             float* __restrict__ den, int N) {
    int idx = blockIdx.x * 256 + threadIdx.x;
    if (idx < N * HD) agg[idx] = 0.0f;
    if (idx < N) { m[idx] = -1.0e9f; den[idx] = 0.0f; }
}

// warp (wave32) per edge: e = leaky_relu(a1.h_src + a2.h_dst + b), seg-max per dst
__global__ __launch_bounds__(256) void edge_score_max(
    const float* __restrict__ h, const float* __restrict__ W_attn,
    const float* __restrict__ b_attn, const int* __restrict__ esrc,
    const int* __restrict__ edst, const int* __restrict__ lvl, int k,
    float* __restrict__ escore, float* __restrict__ mbuf, int E) {
    int e = blockIdx.x * 8 + (threadIdx.x >> 5);
    if (e >= E) return;
    int dst = edst[e];
    if (lvl[dst] != k) return;
    int lane = threadIdx.x & 31;
    int src  = esrc[e];
    float4 hs = ((const float4*)(h + (size_t)src * HD))[lane];
    float4 hd = ((const float4*)(h + (size_t)dst * HD))[lane];
    float4 a1 = ((const float4*)W_attn)[lane];
    float4 a2 = ((const float4*)W_attn)[lane + 32];
    float s = hs.x*a1.x + hs.y*a1.y + hs.z*a1.z + hs.w*a1.w
            + hd.x*a2.x + hd.y*a2.y + hd.z*a2.z + hd.w*a2.w;
    #pragma unroll
    for (int off = 16; off > 0; off >>= 1) s += __shfl_xor(s, off, 32);
    if (lane == 0) {
        s += b_attn[0];
        s = (s > 0.0f) ? s : 0.2f * s;          // leaky_relu slope 0.2
        escore[e] = s;
        atomicMaxFloat(&mbuf[dst], s);
    }
}

__global__ void edge_exp_sum(const int* __restrict__ edst, const int* __restrict__ lvl,
                             int k, float* __restrict__ escore,
                             const float* __restrict__ mbuf, float* __restrict__ den, int E) {
    int e = blockIdx.x * 256 + threadIdx.x;
    if (e >= E) return;
    int dst = edst[e];
    if (lvl[dst] != k) return;
    float ex = expf(escore[e] - mbuf[dst]);
    escore[e] = ex;
    atomicAdd(&den[dst], ex);
}

// warp per edge: agg[dst] += alpha * h[src]   (softmax-weighted aggregate)
__global__ __launch_bounds__(256) void edge_agg(
    const float* __restrict__ h, const int* __restrict__ esrc,
    const int* __restrict__ edst, const int* __restrict__ lvl, int k,
    const float* __restrict__ escore, const float* __restrict__ den,
    float* __restrict__ agg, int E) {
    int e = blockIdx.x * 8 + (threadIdx.x >> 5);
    if (e >= E) return;
    int dst = edst[e];
    if (lvl[dst] != k) return;
    int lane = threadIdx.x & 31;
    int src  = esrc[e];
    float alpha = escore[e] / (den[dst] + 1e-16f);
    float4 hs = ((const float4*)(h + (size_t)src * HD))[lane];
    float* ap = agg + (size_t)dst * HD + lane * 4;
    atomicAdd(ap + 0, alpha * hs.x);
    atomicAdd(ap + 1, alpha * hs.y);
    atomicAdd(ap + 2, alpha * hs.z);
    atomicAdd(ap + 3, alpha * hs.w);
}

// ---------------------------------------------------------------------------
// GRU + message GEMMs via V_WMMA_F32_16X16X4_F32, M=32 nodes per block.
// Each wave reuses every B (weight) fragment for TWO 16-row tiles, halving
// L2 weight traffic vs M=16 and doubling wmma density per load.
//   msg = agg @ W_msg^T + b_msg            (32x128 @ 128x128)
//   gi  = msg @ W_ih[:, :128]^T            (32x128 @ 128x384)
//   gh  = h   @ W_hh^T                     (32x128 @ 128x384)
__global__ __launch_bounds__(256) void gru_wmma(
    float* __restrict__ h, const float* __restrict__ agg,
    const float* __restrict__ node_type, const float* __restrict__ denom,
    const float* __restrict__ W_msg, const float* __restrict__ b_msg,
    const float* __restrict__ W_ih, const float* __restrict__ W_hh,
    const float* __restrict__ b_ih, const float* __restrict__ b_hh, int N) {
    __shared__ float Hsh[MB * AST];     // current h rows          (16.9 KB)
    __shared__ float Ash[MB * AST];     // agg rows -> msg rows    (16.9 KB)
    __shared__ float Ssh[MB * 256];     // gi+gh pre-act, r,z gates (32 KB)
    __shared__ float InnSh[MB * 128];   // gi for n gate            (16 KB)
    __shared__ float HnSh[MB * 128];    // gh for n gate            (16 KB)
    __shared__ float ntSh[MB * 4];
    __shared__ float dSh[MB];

    const int tid  = threadIdx.x;
    const int wave = tid >> 5;
    const int lane = tid & 31;
    const int half = lane >> 4;
    const int lm   = lane & 15;
    const int node0 = blockIdx.x * MB;

    // ---- cooperative loads: 32 rows x 32 float4 (h and agg), 4 per thread
    #pragma unroll
    for (int it = 0; it < 4; ++it) {
        int idx = tid + it * 256;         // 0..1023
        int row = idx >> 5;
        int c4  = idx & 31;
        int node = node0 + row;
        float4 hv = make_float4(0, 0, 0, 0), av = hv;
        if (node < N) {
            hv = ((const float4*)(h   + (size_t)node * HD))[c4];
            av = ((const float4*)(agg + (size_t)node * HD))[c4];
        }
        float* hp = &Hsh[row * AST + c4 * 4];
        hp[0] = hv.x; hp[1] = hv.y; hp[2] = hv.z; hp[3] = hv.w;
        float* ap = &Ash[row * AST + c4 * 4];
        ap[0] = av.x; ap[1] = av.y; ap[2] = av.z; ap[3] = av.w;
    }
    if (tid < MB) {
        int node = node0 + tid;
        dSh[tid] = (node < N) ? denom[node] : 0.0f;
        float n0 = 0.f, n1 = 0.f, n2 = 0.f;
        if (node < N) {
            n0 = node_type[node * 3 + 0];
            n1 = node_type[node * 3 + 1];
            n2 = node_type[node * 3 + 2];
        }
        ntSh[tid * 4 + 0] = n0; ntSh[tid * 4 + 1] = n1; ntSh[tid * 4 + 2] = n2;
    }
    __syncthreads();

    // ---- Stage 1: msg column-tile `wave` (cols j0..j0+15), 2 row tiles
    {
        const int j0 = wave * 16;
        v8f c0 = {}, c1 = {};
        const float* Bp = W_msg + (size_t)(j0 + lm) * HD;   // B[k][n] = W_msg[n][k]
        #pragma unroll 2
        for (int k0 = 0; k0 < HD; k0 += 4) {
            int ka = k0 + 2 * half;
            v2f b;  b.x  = Bp[ka];                    b.y  = Bp[ka + 1];
            v2f a0; a0.x = Ash[lm * AST + ka];        a0.y = Ash[lm * AST + ka + 1];
            v2f a1; a1.x = Ash[(16 + lm) * AST + ka]; a1.y = Ash[(16 + lm) * AST + ka + 1];
            c0 = __builtin_amdgcn_wmma_f32_16x16x4_f32(false, a0, false, b,
                                                       (short)0, c0, false, false);
            c1 = __builtin_amdgcn_wmma_f32_16x16x4_f32(false, a1, false, b,
                                                       (short)0, c1, false, false);
        }
        __syncthreads();                    // everyone done reading agg in Ash
        float bias = b_msg[j0 + lm];
        #pragma unroll
        for (int r = 0; r < 8; ++r) {
            int row = half ? (r + 8) : r;   // C/D: M = r + 8*half, N = lm
            Ash[row * AST + j0 + lm]        = c0[r] + bias;
            Ash[(16 + row) * AST + j0 + lm] = c1[r] + bias;
        }
    }
    __syncthreads();

    // ---- Stage 2: 24 gate column-blocks (384 cols), 3 per wave; gi & gh,
    //      each B fragment reused across both 16-row tiles
    #pragma unroll
    for (int tb = 0; tb < 3; ++tb) {
        const int t   = wave + tb * 8;      // 0..23
        const int jj0 = t * 16;             // gate-row block in [0,384)
        v8f cgi0 = {}, cgi1 = {}, cgh0 = {}, cgh1 = {};
        const float* Bi = W_ih + (size_t)(jj0 + lm) * (HD + 3);
        const float* Bh = W_hh + (size_t)(jj0 + lm) * HD;
        #pragma unroll 2
        for (int k0 = 0; k0 < HD; k0 += 4) {
            int ka = k0 + 2 * half;
            v2f am0; am0.x = Ash[lm * AST + ka];        am0.y = Ash[lm * AST + ka + 1];
            v2f am1; am1.x = Ash[(16 + lm) * AST + ka]; am1.y = Ash[(16 + lm) * AST + ka + 1];
            v2f ah0; ah0.x = Hsh[lm * AST + ka];        ah0.y = Hsh[lm * AST + ka + 1];
            v2f ah1; ah1.x = Hsh[(16 + lm) * AST + ka]; ah1.y = Hsh[(16 + lm) * AST + ka + 1];
            v2f bi; bi.x = Bi[ka]; bi.y = Bi[ka + 1];
            v2f bh; bh.x = Bh[ka]; bh.y = Bh[ka + 1];
            cgi0 = __builtin_amdgcn_wmma_f32_16x16x4_f32(false, am0, false, bi,
                                                         (short)0, cgi0, false, false);
            cgi1 = __builtin_amdgcn_wmma_f32_16x16x4_f32(false, am1, false, bi,
                                                         (short)0, cgi1, false, false);
            cgh0 = __builtin_amdgcn_wmma_f32_16x16x4_f32(false, ah0, false, bh,
                                                         (short)0, cgh0, false, false);
            cgh1 = __builtin_amdgcn_wmma_f32_16x16x4_f32(false, ah1, false, bh,
                                                         (short)0, cgh1, false, false);
        }
        if (t < 16) {            // r,z gates only need the sum gi+gh
            #pragma unroll
            for (int r = 0; r < 8; ++r) {
                int row = half ? (r + 8) : r;
                Ssh[row * 256 + jj0 + lm]        = cgi0[r] + cgh0[r];
                Ssh[(16 + row) * 256 + jj0 + lm] = cgi1[r] + cgh1[r];
            }
        } else {                 // n gate needs gi and gh separately (r * hn)
            int jn = jj0 - 256;
            #pragma unroll
            for (int r = 0; r < 8; ++r) {
                int row = half ? (r + 8) : r;
                InnSh[row * 128 + jn + lm]        = cgi0[r];
                HnSh [row * 128 + jn + lm]        = cgh0[r];
                InnSh[(16 + row) * 128 + jn + lm] = cgi1[r];
                HnSh [(16 + row) * 128 + jn + lm] = cgh1[r];
            }
        }
    }
    __syncthreads();

    // ---- epilogue: node_type contribution + biases + gates + masked update
    const int i  = tid >> 3;                // node row 0..31
    const int j0 = (tid & 7) * 16;          // 16 hidden cols per thread
    const int node = node0 + i;
    const bool active = (node < N) && (dSh[i] > 0.0f);
    const float nt0 = ntSh[i * 4 + 0], nt1 = ntSh[i * 4 + 1], nt2 = ntSh[i * 4 + 2];
    #pragma unroll 4
    for (int jc = 0; jc < 16; ++jc) {
        int j = j0 + jc;
        const float* wr = W_ih + (size_t)j         * (HD + 3) + HD;
        const float* wz = W_ih + (size_t)(128 + j) * (HD + 3) + HD;
        const float* wn = W_ih + (size_t)(256 + j) * (HD + 3) + HD;
        float ntc_r = nt0 * wr[0] + nt1 * wr[1] + nt2 * wr[2];
        float ntc_z = nt0 * wz[0] + nt1 * wz[1] + nt2 * wz[2];
        float ntc_n = nt0 * wn[0] + nt1 * wn[1] + nt2 * wn[2];
        float rr = 1.0f / (1.0f + expf(-(Ssh[i * 256 + j]       + b_ih[j]       + b_hh[j]       + ntc_r)));
        float zz = 1.0f / (1.0f + expf(-(Ssh[i * 256 + 128 + j] + b_ih[128 + j] + b_hh[128 + j] + ntc_z)));
        float nn = tanhf(InnSh[i * 128 + j] + b_ih[256 + j] + ntc_n
                         + rr * (HnSh[i * 128 + j] + b_hh[256 + j]));
        float hold = Hsh[i * AST + j];
        float hnew = (1.0f - zz) * nn + zz * hold;
        if (active) h[(size_t)node * HD + j] = hnew;
    }
}

// ---------------------------------------------------------------------------
extern "C" void kernel_launch(void* const* d_in, const int* in_sizes, int n_in,
                              void* d_out, int out_size, void* d_ws, size_t ws_size,
                              hipStream_t stream) {
    const float* node_embedding = (const float*)d_in[0];
    const float* node_type      = (const float*)d_in[1];
    const float* W_attn         = (const float*)d_in[2];
    const float* b_attn         = (const float*)d_in[3];
    const float* W_msg          = (const float*)d_in[4];
    const float* b_msg          = (const float*)d_in[5];
    const float* W_ih           = (const float*)d_in[6];
    const float* W_hh           = (const float*)d_in[7];
    const float* b_ih           = (const float*)d_in[8];
    const float* b_hh           = (const float*)d_in[9];
    const int*   edge_src       = (const int*)d_in[10];
    const int*   edge_dst       = (const int*)d_in[11];
    const int*   lvl            = (const int*)d_in[12];

    const int N = in_sizes[0] / HD;
    const int E = in_sizes[10];

    float* h      = (float*)d_out;
    float* escore = (float*)d_ws;          // [E]
    float* mbuf   = escore + E;            // [N]
    float* den    = mbuf + N;              // [N]
    float* agg    = den + N;               // [N*HD]

    // h starts as node_embedding (do not mutate inputs)
    hipMemcpyAsync(h, node_embedding, (size_t)N * HD * sizeof(float),
                   hipMemcpyDeviceToDevice, stream);

    const int rstBlocks  = (N * HD + 255) / 256;
    const int warpBlocks = (E + 7) / 8;     // 8 wave32 warps per 256-thread block
    const int thrBlocks  = (E + 255) / 256;
    const int gruBlocks  = (N + MB - 1) / MB;

    for (int k = 1; k < DEPTH; ++k) {       // sequential DAG level propagation
        reset_kernel<<<rstBlocks, 256, 0, stream>>>(agg, mbuf, den, N);
        edge_score_max<<<warpBlocks, 256, 0, stream>>>(h, W_attn, b_attn,
                edge_src, edge_dst, lvl, k, escore, mbuf, E);
        edge_exp_sum<<<thrBlocks, 256, 0, stream>>>(edge_dst, lvl, k,
                escore, mbuf, den, E);
        edge_agg<<<warpBlocks, 256, 0, stream>>>(h, edge_src, edge_dst, lvl, k,
                escore, den, agg, E);
        gru_wmma<<<gruBlocks, 256, 0, stream>>>(h, agg, node_type, den,
                W_msg, b_msg, W_ih, W_hh, b_ih, b_hh, N);
    }
}